// AE_LSTM_34857954574922
// MI455X (gfx1250) — compile-verified
//
#include <hip/hip_runtime.h>
#include <hip/hip_bf16.h>

// ---------------------------------------------------------------------------
// AE-LSTM for MI455X (gfx1250). All GEMMs run on the WMMA pipe using the
// exact-fp32 V_WMMA_F32_16X16X4_F32 instruction. B-fragments are prefetched
// into register arrays so global loads issue as clauses and overlap with the
// WMMA chain instead of serializing on s_wait_loadcnt 0.
// ---------------------------------------------------------------------------

typedef __attribute__((ext_vector_type(2))) float v2f;
typedef __attribute__((ext_vector_type(8))) float v8f;

#define NB    256   // batch
#define NF    81    // freq
#define NT    100   // seq len
#define NH    128   // hidden
#define TLAG  30
#define NKEEP 40
#define NN    (NB * NKEEP)   // 10240 decoder rows

__device__ __forceinline__ v8f wmma4(v2f a, v2f b, v8f c) {
    // D = A(16x4 f32) * B(4x16 f32) + C(16x16 f32)  -- exact fp32
    return __builtin_amdgcn_wmma_f32_16x16x4_f32(
        false, a, false, b, (short)0, c, false, false);
}

__device__ __forceinline__ float sigf(float x) {
    return 1.0f / (1.0f + __expf(-x));
}

// ===========================================================================
// Kernel 1: conv encoder (3x Conv2d(k=(3,1)) + 2x MaxPool(2,1)) + dense 640->40
// One block of 128 threads per (b,t) column; freq-only convs so columns are
// independent. Output e[B,T,40].
// ===========================================================================
__global__ __launch_bounds__(128)
void ae_enc_conv_kernel(const float* __restrict__ x,
                        const float* __restrict__ w1, const float* __restrict__ b1,
                        const float* __restrict__ w2, const float* __restrict__ b2,
                        const float* __restrict__ w3, const float* __restrict__ b3,
                        const float* __restrict__ We, const float* __restrict__ be,
                        float* __restrict__ e) {
    __shared__ float xin[NF];
    __shared__ float t1[8 * NF];     // conv1 out
    __shared__ float s1[8 * 40];     // pool1
    __shared__ float t2[16 * 40];    // conv2 out
    __shared__ float s2[16 * 20];    // pool2
    __shared__ float t3[32 * 20];    // conv3 out
    const int tid = threadIdx.x;
    const int b = blockIdx.x / NT;
    const int t = blockIdx.x % NT;

    for (int f = tid; f < NF; f += 128) xin[f] = x[b * (NF * NT) + f * NT + t];
    __syncthreads();

    for (int i = tid; i < 8 * NF; i += 128) {
        int c = i / NF, f = i % NF;
        float acc = b1[c];
#pragma unroll
        for (int k = 0; k < 3; ++k) {
            int ff = f + k - 1;
            if ((unsigned)ff < (unsigned)NF) acc += w1[c * 3 + k] * xin[ff];
        }
        t1[i] = fmaxf(acc, 0.0f);
    }
    __syncthreads();
    for (int i = tid; i < 8 * 40; i += 128) {
        int c = i / 40, p = i % 40;
        s1[i] = fmaxf(t1[c * NF + 2 * p], t1[c * NF + 2 * p + 1]);
    }
    __syncthreads();
    for (int i = tid; i < 16 * 40; i += 128) {
        int c = i / 40, f = i % 40;
        float acc = b2[c];
        for (int ci = 0; ci < 8; ++ci)
#pragma unroll
            for (int k = 0; k < 3; ++k) {
                int ff = f + k - 1;
                if ((unsigned)ff < 40u) acc += w2[(c * 8 + ci) * 3 + k] * s1[ci * 40 + ff];
            }
        t2[i] = fmaxf(acc, 0.0f);
    }
    __syncthreads();
    for (int i = tid; i < 16 * 20; i += 128) {
        int c = i / 20, p = i % 20;
        s2[i] = fmaxf(t2[c * 40 + 2 * p], t2[c * 40 + 2 * p + 1]);
    }
    __syncthreads();
    for (int i = tid; i < 32 * 20; i += 128) {
        int c = i / 20, f = i % 20;
        float acc = b3[c];
        for (int ci = 0; ci < 16; ++ci)
#pragma unroll
            for (int k = 0; k < 3; ++k) {
                int ff = f + k - 1;
                if ((unsigned)ff < 20u) acc += w3[(c * 16 + ci) * 3 + k] * s2[ci * 20 + ff];
            }
        t3[i] = fmaxf(acc, 0.0f);
    }
    __syncthreads();
    // dense 640->40; feature index = h*32 + c (from transpose(0,3,2,1))
    for (int j = tid; j < 40; j += 128) {
        float acc = be[j];
        for (int h = 0; h < 20; ++h)
            for (int c = 0; c < 32; ++c)
                acc += t3[c * 20 + h] * We[j * 640 + h * 32 + c];
        e[(b * NT + t) * 40 + j] = acc;
    }
}

// ===========================================================================
// Kernel 2: encoder LSTM. 16 WGs x 16 batch rows, 256 thr (8 waves).
// gates[16,512] = x[16,40] @ Wih^T + h[16,128] @ Whh^T + bias via fp32 WMMA.
// A-fragments pre-staged per step; B-fragments prefetched per tile so the
// weight loads issue as a clause and overlap the WMMA chain.
// ===========================================================================
__global__ __launch_bounds__(256)
void ae_enc_lstm_kernel(const float* __restrict__ e,
                        const float* __restrict__ Wih, const float* __restrict__ Whh,
                        const float* __restrict__ bih, const float* __restrict__ bhh,
                        float* __restrict__ hk, float* __restrict__ ck) {
    __shared__ float xt[16 * 40];
    __shared__ float hbuf[16 * NH];
    __shared__ float cbuf[16 * NH];
    __shared__ float gbuf[16 * 512];

    const int tid = threadIdx.x;
    const int wave = tid >> 5;
    const int lane = tid & 31;
    const int b0 = blockIdx.x * 16;
    const int mrow = lane & 15;            // A row / B col / C col
    const int ko = (lane >> 4) * 2;        // K sub-offset for lanes 16-31
    const int mbase = (lane >> 4) * 8;     // C row base

    for (int i = tid; i < 16 * NH; i += 256) { hbuf[i] = 0.0f; cbuf[i] = 0.0f; }
    __syncthreads();

    for (int t = 0; t < NT; ++t) {
        for (int i = tid; i < 16 * 40; i += 256) {
            int r = i / 40, j = i % 40;
            xt[i] = e[((b0 + r) * NT + t) * 40 + j];
        }
        __syncthreads();

        // Pre-stage A-fragments (shared across all N-tiles of this step)
        v2f ax[10], ah[32];
#pragma unroll
        for (int kk = 0; kk < 10; ++kk)
            ax[kk] = *(const v2f*)&xt[mrow * 40 + kk * 4 + ko];
#pragma unroll
        for (int kk = 0; kk < 32; ++kk)
            ah[kk] = *(const v2f*)&hbuf[mrow * NH + kk * 4 + ko];

        // 32 N-tiles of 16 gate columns; 4 tiles per wave
        for (int nt = wave; nt < 32; nt += 8) {
            const int n0 = nt * 16;
            // Prefetch B-fragments into registers (distinct dests -> load clause)
            v2f bx[10], bh[32];
            const float* wpx = Wih + (n0 + mrow) * 40 + ko;
            const float* wph = Whh + (n0 + mrow) * NH + ko;
#pragma unroll
            for (int kk = 0; kk < 10; ++kk) bx[kk] = *(const v2f*)(wpx + kk * 4);
#pragma unroll
            for (int kk = 0; kk < 32; ++kk) bh[kk] = *(const v2f*)(wph + kk * 4);

            v8f acc = {0.f, 0.f, 0.f, 0.f, 0.f, 0.f, 0.f, 0.f};
#pragma unroll
            for (int kk = 0; kk < 10; ++kk) acc = wmma4(ax[kk], bx[kk], acc);
#pragma unroll
            for (int kk = 0; kk < 32; ++kk) acc = wmma4(ah[kk], bh[kk], acc);

            const float bias = bih[n0 + mrow] + bhh[n0 + mrow];
#pragma unroll
            for (int v = 0; v < 8; ++v)
                gbuf[(mbase + v) * 512 + n0 + mrow] = acc[v] + bias;
        }
        __syncthreads();

        // activations + state update (i,f,g,o gate order)
        for (int i = tid; i < 16 * NH; i += 256) {
            const int r = i >> 7, hc = i & 127;
            const float ig = gbuf[r * 512 + hc];
            const float fg = gbuf[r * 512 + 128 + hc];
            const float gg = gbuf[r * 512 + 256 + hc];
            const float og = gbuf[r * 512 + 384 + hc];
            float c = sigf(fg) * cbuf[i] + sigf(ig) * tanhf(gg);
            float h = sigf(og) * tanhf(c);
            cbuf[i] = c; hbuf[i] = h;
            if (t >= TLAG + 30) {  // keep t >= 60
                const int n = (b0 + r) * NKEEP + (t - 60);
                hk[n * NH + hc] = h;
                ck[n * NH + hc] = c;
            }
        }
        __syncthreads();
    }
}

// ===========================================================================
// Kernel 3: fused decoder. 640 WGs x 16 rows, 256 thr.
// Per step: WMMA gates (x=0 so only h@Whh^T + bias) -> activations ->
// WMMA dense 128->320 -> convT chain (VALU in LDS) -> store outputs.
// Hidden states never touch HBM (~315 MB saved).
// ===========================================================================
__global__ __launch_bounds__(256)
void ae_dec_fused_kernel(const float* __restrict__ hk, const float* __restrict__ ck,
                         const float* __restrict__ Whh,
                         const float* __restrict__ bih, const float* __restrict__ bhh,
                         const float* __restrict__ Wd, const float* __restrict__ bd,
                         const float* __restrict__ wt1, const float* __restrict__ bt1,
                         const float* __restrict__ wt2, const float* __restrict__ bt2,
                         const float* __restrict__ wt3, const float* __restrict__ bt3,
                         float* __restrict__ out) {
    extern __shared__ float smem[];
    float* gbias = smem;                  // 512
    float* hbuf  = gbias + 512;           // 16*128
    float* cbuf  = hbuf + 16 * NH;        // 16*128
    float* gbuf  = cbuf + 16 * NH;        // 16*512 (gates; reused as z[16,320] row-stride 512)
    float* o1    = gbuf + 16 * 512;       // 16*320 (convT1 out)
    float* o2    = o1 + 16 * 320;         // 16*328 (convT2 out)

    const int tid = threadIdx.x;
    const int wave = tid >> 5;
    const int lane = tid & 31;
    const int n0 = blockIdx.x * 16;
    const int mrow = lane & 15;
    const int ko = (lane >> 4) * 2;
    const int mbase = (lane >> 4) * 8;

    for (int i = tid; i < 512; i += 256) gbias[i] = bih[i] + bhh[i];
    for (int i = tid; i < 16 * NH; i += 256) {
        const int r = i >> 7, j = i & 127;
        hbuf[i] = hk[(n0 + r) * NH + j];
        cbuf[i] = ck[(n0 + r) * NH + j];
    }
    __syncthreads();

    for (int tl = 0; tl < TLAG; ++tl) {
        // ---- gates: [16,128] @ Whh^T -> [16,512]  (x == 0, so no Wih term)
        v2f ah[32];
#pragma unroll
        for (int kk = 0; kk < 32; ++kk)
            ah[kk] = *(const v2f*)&hbuf[mrow * NH + kk * 4 + ko];

        for (int nt = wave; nt < 32; nt += 8) {
            const int g0 = nt * 16;
            v2f bh[32];
            const float* wph = Whh + (g0 + mrow) * NH + ko;
#pragma unroll
            for (int kk = 0; kk < 32; ++kk) bh[kk] = *(const v2f*)(wph + kk * 4);

            v8f acc = {0.f, 0.f, 0.f, 0.f, 0.f, 0.f, 0.f, 0.f};
#pragma unroll
            for (int kk = 0; kk < 32; ++kk) acc = wmma4(ah[kk], bh[kk], acc);

            const float bias = gbias[g0 + mrow];
#pragma unroll
            for (int v = 0; v < 8; ++v)
                gbuf[(mbase + v) * 512 + g0 + mrow] = acc[v] + bias;
        }
        __syncthreads();

        for (int i = tid; i < 16 * NH; i += 256) {
            const int r = i >> 7, hc = i & 127;
            const float ig = gbuf[r * 512 + hc];
            const float fg = gbuf[r * 512 + 128 + hc];
            const float gg = gbuf[r * 512 + 256 + hc];
            const float og = gbuf[r * 512 + 384 + hc];
            float c = sigf(fg) * cbuf[i] + sigf(ig) * tanhf(gg);
            float h = sigf(og) * tanhf(c);
            cbuf[i] = c; hbuf[i] = h;
        }
        __syncthreads();

        // ---- dense: z[16,320] = h_new @ Wd^T + bd (stored in gbuf, stride 512)
        v2f az[32];
#pragma unroll
        for (int kk = 0; kk < 32; ++kk)
            az[kk] = *(const v2f*)&hbuf[mrow * NH + kk * 4 + ko];

        for (int nt = wave; nt < 20; nt += 8) {
            const int z0 = nt * 16;
            v2f bz[32];
            const float* wpz = Wd + (z0 + mrow) * NH + ko;
#pragma unroll
            for (int kk = 0; kk < 32; ++kk) bz[kk] = *(const v2f*)(wpz + kk * 4);

            v8f acc = {0.f, 0.f, 0.f, 0.f, 0.f, 0.f, 0.f, 0.f};
#pragma unroll
            for (int kk = 0; kk < 32; ++kk) acc = wmma4(az[kk], bz[kk], acc);

            const float bias = bd[z0 + mrow];
#pragma unroll
            for (int v = 0; v < 8; ++v)
                gbuf[(mbase + v) * 512 + z0 + mrow] = acc[v] + bias;
        }
        __syncthreads();

        // ---- convT1: [32,10] -> relu -> [16,20], stride 2, pad 1, opad 1
        for (int i = tid; i < 16 * 16 * 20; i += 256) {
            const int s = i / 320, rem = i % 320, co = rem / 20, tp = rem % 20;
            float acc = bt1[co];
#pragma unroll
            for (int k = 0; k < 3; ++k) {
                const int num = tp + 1 - k;
                if (num >= 0 && !(num & 1)) {
                    const int sp = num >> 1;
                    if (sp < 10)
                        for (int ci = 0; ci < 32; ++ci)
                            acc += gbuf[s * 512 + ci * 10 + sp] * wt1[(ci * 16 + co) * 3 + k];
                }
            }
            o1[s * 320 + co * 20 + tp] = fmaxf(acc, 0.0f);
        }
        __syncthreads();

        // ---- convT2: [16,20] -> relu -> [8,41], stride 2, pad 0
        for (int i = tid; i < 16 * 8 * 41; i += 256) {
            const int s = i / 328, rem = i % 328, co = rem / 41, tp = rem % 41;
            float acc = bt2[co];
#pragma unroll
            for (int k = 0; k < 3; ++k) {
                const int num = tp - k;
                if (num >= 0 && !(num & 1)) {
                    const int sp = num >> 1;
                    if (sp < 20)
                        for (int ci = 0; ci < 16; ++ci)
                            acc += o1[s * 320 + ci * 20 + sp] * wt2[(ci * 8 + co) * 3 + k];
                }
            }
            o2[s * 328 + co * 41 + tp] = fmaxf(acc, 0.0f);
        }
        __syncthreads();

        // ---- convT3: [8,41] -> [1,81], stride 2, pad 1; write output
        for (int i = tid; i < 16 * 81; i += 256) {
            const int s = i / 81, tp = i % 81;
            float acc = bt3[0];
#pragma unroll
            for (int k = 0; k < 3; ++k) {
                const int num = tp + 1 - k;
                if (num >= 0 && !(num & 1)) {
                    const int sp = num >> 1;
                    if (sp < 41)
                        for (int ci = 0; ci < 8; ++ci)
                            acc += o2[s * 328 + ci * 41 + sp] * wt3[ci * 3 + k];
                }
            }
            // out[b,keep,tl,f] with n = b*KEEP+keep -> ((n*TL)+tl)*F + f
            out[((n0 + s) * TLAG + tl) * NF + tp] = acc;
        }
        __syncthreads();
    }
}

// ===========================================================================
extern "C" void kernel_launch(void* const* d_in, const int* in_sizes, int n_in,
                              void* d_out, int out_size, void* d_ws, size_t ws_size,
                              hipStream_t stream) {
    const float* x     = (const float*)d_in[0];
    const float* w1    = (const float*)d_in[1];
    const float* b1    = (const float*)d_in[2];
    const float* w2    = (const float*)d_in[3];
    const float* b2    = (const float*)d_in[4];
    const float* w3    = (const float*)d_in[5];
    const float* b3    = (const float*)d_in[6];
    const float* We    = (const float*)d_in[7];
    const float* be    = (const float*)d_in[8];
    const float* Wih_e = (const float*)d_in[9];
    const float* Whh_e = (const float*)d_in[10];
    const float* bih_e = (const float*)d_in[11];
    const float* bhh_e = (const float*)d_in[12];
    // d_in[13] = Wih_d: decoder input is identically zero, so x@Wih_d^T == 0.
    const float* Whh_d = (const float*)d_in[14];
    const float* bih_d = (const float*)d_in[15];
    const float* bhh_d = (const float*)d_in[16];
    const float* Wd    = (const float*)d_in[17];
    const float* bd    = (const float*)d_in[18];
    const float* wt1   = (const float*)d_in[19];
    const float* bt1   = (const float*)d_in[20];
    const float* wt2   = (const float*)d_in[21];
    const float* bt2   = (const float*)d_in[22];
    const float* wt3   = (const float*)d_in[23];
    const float* bt3   = (const float*)d_in[24];
    float* out = (float*)d_out;

    // workspace: e[B,T,40] | hk[N,128] | ck[N,128]  (~14.6 MB)
    float* ws = (float*)d_ws;
    float* e  = ws;
    float* hk = e + (size_t)NB * NT * 40;
    float* ck = hk + (size_t)NN * NH;

    ae_enc_conv_kernel<<<dim3(NB * NT), dim3(128), 0, stream>>>(
        x, w1, b1, w2, b2, w3, b3, We, be, e);

    ae_enc_lstm_kernel<<<dim3(NB / 16), dim3(256), 0, stream>>>(
        e, Wih_e, Whh_e, bih_e, bhh_e, hk, ck);

    const size_t smemBytes =
        (512 + 2 * 16 * NH + 16 * 512 + 16 * 320 + 16 * 328) * sizeof(float); // ~92.7 KB
    ae_dec_fused_kernel<<<dim3(NN / 16), dim3(256), smemBytes, stream>>>(
        hk, ck, Whh_d, bih_d, bhh_d, Wd, bd, wt1, bt1, wt2, bt2, wt3, bt3, out);
}